// MyModel_87522843560689
// MI455X (gfx1250) — compile-verified
//
#include <hip/hip_runtime.h>

// RBF-PINN derivative evaluation, fused for MI455X (gfx1250, wave32, WMMA).
//
// G[i,j] = exp(-0.5*((cx-x)^2+(cz-z)^2+(ct-t)^2))
// All 17 outputs = linear combos of Y = G @ S, where S[:, col] are v_k
// scaled by {1, cx, cz, ct, cx^2, cz^2} (24 cols, padded to 32).
// WMMA: V_WMMA_F32_16X16X4_F32, A = fresh 16x4 f32 G tile (2 exps/lane/step).
// S is stored COLUMN-major so each B fragment is one aligned b64 load;
// centers are repacked to float4 so each G step needs two b128 loads.

#define NPTS 6144
#define SCOLS 32
#define ROWS 16          // WMMA M tile
#define WAVES 8          // waves per block (256 threads, wave32)
#define KSLICE (NPTS / WAVES)   // 768 K per wave

typedef __attribute__((ext_vector_type(2))) float v2f;
typedef __attribute__((ext_vector_type(8))) float v8f;

// ---------------------------------------------------------------- kernel 1 --
// ST[col*NPTS + j]: per RHS group g (0=v1,1=v2,2=v3,3=v4) at col base g*6:
//   {v, v*cx, v*cz, v*ct, v*cx^2, v*cz^2}; cols 24..31 zero padding.
// cp4[j] = {cx, cz, ct, 0} (16B-aligned center rows).
__global__ void build_S_kernel(const float* __restrict__ c,
                               const float* __restrict__ v1,
                               const float* __restrict__ v2,
                               const float* __restrict__ v3,
                               const float* __restrict__ v4,
                               float* __restrict__ ST,
                               float4* __restrict__ cp4) {
    int j = blockIdx.x * blockDim.x + threadIdx.x;
    if (j >= NPTS) return;
    float cx = c[3 * j + 0], cz = c[3 * j + 1], ct = c[3 * j + 2];
    cp4[j] = make_float4(cx, cz, ct, 0.0f);
    float vs[4] = { v1[j], v2[j], v3[j], v4[j] };
#pragma unroll
    for (int g = 0; g < 4; ++g) {
        float v = vs[g];
        ST[(g * 6 + 0) * NPTS + j] = v;
        ST[(g * 6 + 1) * NPTS + j] = v * cx;
        ST[(g * 6 + 2) * NPTS + j] = v * cz;
        ST[(g * 6 + 3) * NPTS + j] = v * ct;
        ST[(g * 6 + 4) * NPTS + j] = v * cx * cx;
        ST[(g * 6 + 5) * NPTS + j] = v * cz * cz;
    }
#pragma unroll
    for (int p = 24; p < 32; ++p) ST[p * NPTS + j] = 0.0f;
}

// ---------------------------------------------------------------- kernel 2 --
// One block = one 16-row strip of outputs. 8 waves split the K=6144 reduction,
// each accumulating two 16x16 f32 WMMA tiles (cols 0-15 and 16-31 of Y).
__global__ __launch_bounds__(256) void rbf_pinn_kernel(
        const float* __restrict__ x, const float* __restrict__ z,
        const float* __restrict__ t, const float4* __restrict__ cp4,
        const float* __restrict__ ST, float* __restrict__ out) {
    __shared__ float part[WAVES * 2 * 256];   // per-wave C tiles (16 KB)
    __shared__ float Y[ROWS * SCOLS];         // reduced Y strip (2 KB)

    const int tid   = threadIdx.x;
    const int lane  = tid & 31;
    const int wave  = tid >> 5;
    const int rbase = blockIdx.x * ROWS;

    // A-fragment mapping for 16x16x4 f32 (ISA 7.12.2): lanes 0-15 hold K=0,1
    // for M=lane; lanes 16-31 hold K=2,3 for M=lane-16. B mirrors with N=lane&15.
    const int m    = lane & 15;
    const int jadd = (lane >> 4) * 2;
    const int i    = rbase + m;

    const float xi = x[i], zi = z[i], ti = t[i];

    const float* STr0 = ST + (size_t)m * NPTS;          // B cols 0-15 stream
    const float* STr1 = ST + (size_t)(m + 16) * NPTS;   // B cols 16-31 stream

    v8f acc0 = {};
    v8f acc1 = {};

    const int j0beg = wave * KSLICE;
    const int j0end = j0beg + KSLICE;

#pragma unroll 2
    for (int j0 = j0beg; j0 < j0end; j0 += 4) {
        const int ja = j0 + jadd;

        // two Gaussian Gram entries -> f32 A fragment (2x b128 center loads)
        float4 c0 = cp4[ja];
        float4 c1 = cp4[ja + 1];
        float dx0 = c0.x - xi, dz0 = c0.y - zi, dt0 = c0.z - ti;
        float dx1 = c1.x - xi, dz1 = c1.y - zi, dt1 = c1.z - ti;
        float r20 = dx0 * dx0 + dz0 * dz0 + dt0 * dt0;
        float r21 = dx1 * dx1 + dz1 * dz1 + dt1 * dt1;
        v2f a = { __expf(-0.5f * r20), __expf(-0.5f * r21) };

        // B fragments: one aligned b64 load per tile (ja is even)
        v2f b0 = *(const v2f*)(STr0 + ja);
        v2f b1 = *(const v2f*)(STr1 + ja);
        __builtin_prefetch(STr0 + ja + 8, 0, 3);

        acc0 = __builtin_amdgcn_wmma_f32_16x16x4_f32(
            false, a, false, b0, (short)0, acc0, false, false);
        acc1 = __builtin_amdgcn_wmma_f32_16x16x4_f32(
            false, a, false, b1, (short)0, acc1, false, false);
    }

    // ---- cross-wave reduction (fixed order -> deterministic) ----
    {
        float* p0 = part + (wave * 2 + 0) * 256;
        float* p1 = part + (wave * 2 + 1) * 256;
#pragma unroll
        for (int r = 0; r < 8; ++r) {
            p0[r * 32 + lane] = acc0[r];
            p1[r * 32 + lane] = acc1[r];
        }
    }
    __syncthreads();

    for (int p = tid; p < 512; p += 256) {
        float s = 0.0f;
#pragma unroll
        for (int w = 0; w < WAVES; ++w) s += part[w * 512 + p];
        // decode C-fragment position: VGPR r, lane l -> row r + 8*(l>>4), col l&15
        int l2  = p & 31;
        int r   = (p >> 5) & 7;
        int col = ((p >> 8) << 4) + (l2 & 15);
        int row = r + 8 * (l2 >> 4);
        Y[row * SCOLS + col] = s;
    }
    __syncthreads();

    // ---- recombine Y into the 17 outputs (16 lanes, one row each) ----
    if (tid < ROWS) {
        const int   gi = rbase + tid;
        const float xr = x[gi], zr = z[gi], tr = t[gi];
        const float* y = Y + tid * SCOLS;

        float p0 = y[0],  px = y[1],  pz = y[2];                       // v1
        float b0 = y[6],  bx = y[7],  bz = y[8],  bt = y[9],
              bxx = y[10], bzz = y[11];                                // v2
        float u0 = y[12], ux = y[13], uz = y[14], ut = y[15],
              uxx = y[16], uzz = y[17];                                // v3
        float w0 = y[18], wx = y[19], wz = y[20], wt = y[21],
              wxx = y[22], wzz = y[23];                                // v4

        float x2m1 = xr * xr - 1.0f, z2m1 = zr * zr - 1.0f;

        out[ 0 * NPTS + gi] = ux - xr * u0;                            // dudx
        out[ 1 * NPTS + gi] = uz - zr * u0;                            // dudz
        out[ 2 * NPTS + gi] = ut - tr * u0;                            // dudt
        out[ 3 * NPTS + gi] = wx - xr * w0;                            // dwdx
        out[ 4 * NPTS + gi] = wz - zr * w0;                            // dwdz
        out[ 5 * NPTS + gi] = wt - tr * w0;                            // dwdt
        out[ 6 * NPTS + gi] = bx - xr * b0;                            // dbdx
        out[ 7 * NPTS + gi] = bz - zr * b0;                            // dbdz
        out[ 8 * NPTS + gi] = bt - tr * b0;                            // dbdt
        out[ 9 * NPTS + gi] = px - xr * p0;                            // dpdx
        out[10 * NPTS + gi] = pz - zr * p0;                            // dpdz
        out[11 * NPTS + gi] = uxx - 2.0f * xr * ux + x2m1 * u0;        // d2ud2x
        out[12 * NPTS + gi] = uzz - 2.0f * zr * uz + z2m1 * u0;        // d2ud2z
        out[13 * NPTS + gi] = wxx - 2.0f * xr * wx + x2m1 * w0;        // d2wd2x
        out[14 * NPTS + gi] = wzz - 2.0f * zr * wz + z2m1 * w0;        // d2wd2z
        out[15 * NPTS + gi] = bxx - 2.0f * xr * bx + x2m1 * b0;        // d2bd2x
        out[16 * NPTS + gi] = bzz - 2.0f * zr * bz + z2m1 * b0;        // d2bd2z
    }
}

// ----------------------------------------------------------------- launcher --
extern "C" void kernel_launch(void* const* d_in, const int* in_sizes, int n_in,
                              void* d_out, int out_size, void* d_ws, size_t ws_size,
                              hipStream_t stream) {
    (void)in_sizes; (void)n_in; (void)out_size; (void)ws_size;
    const float* x  = (const float*)d_in[0];
    const float* z  = (const float*)d_in[1];
    const float* t  = (const float*)d_in[2];
    const float* c  = (const float*)d_in[3];
    const float* v1 = (const float*)d_in[4];
    const float* v2 = (const float*)d_in[5];
    const float* v3 = (const float*)d_in[6];
    const float* v4 = (const float*)d_in[7];
    float* out = (float*)d_out;

    // workspace layout: ST (column-major S) then float4 centers
    float*  ST  = (float*)d_ws;                                   // 786,432 B
    float4* cp4 = (float4*)((char*)d_ws + (size_t)NPTS * SCOLS * 4); // +98,304 B

    build_S_kernel<<<NPTS / 256, 256, 0, stream>>>(c, v1, v2, v3, v4, ST, cp4);
    rbf_pinn_kernel<<<NPTS / ROWS, 256, 0, stream>>>(x, z, t, cp4, ST, out);
}